// dcrnn_81647328297651
// MI455X (gfx1250) — compile-verified
//
#include <hip/hip_runtime.h>

typedef __attribute__((ext_vector_type(2))) float v2f;
typedef __attribute__((ext_vector_type(8))) float v8f;

#define NN   50000
#define NE   1600000
#define CIN  32
#define CHID 64
#define NTILES (NN / 16)   // 3125 exactly

// ---------------------------------------------------------------- zero scratch
__global__ void zero_ws(float* __restrict__ p, int n) {
  int i = blockIdx.x * blockDim.x + threadIdx.x;
  if (i < n) p[i] = 0.f;
}

// ------------------------------------------------- pack weights (rows 0..31 only)
// W layout [2,K=2,96,64]; H0==0 => only rows 0..31 of the 96-row dim matter.
// pw layout (floats): gate-z {k0sum[2048], W01[2048], W11[2048]}, then gate-h same.
__global__ void pack_w(const float* __restrict__ Wz, const float* __restrict__ Wh,
                       float* __restrict__ pw) {
  int t = blockIdx.x * blockDim.x + threadIdx.x;
  if (t >= 3 * 2048) return;
  int m = t >> 11;          // 0: k0 sum, 1: W[0,1], 2: W[1,1]
  int i = t & 2047;         // k*64 + n, k<32
  const int s00 = 0, s01 = 96 * 64, s10 = 2 * 96 * 64, s11 = 3 * 96 * 64;
  float vz, vh;
  if (m == 0)      { vz = Wz[s00 + i] + Wz[s10 + i]; vh = Wh[s00 + i] + Wh[s10 + i]; }
  else if (m == 1) { vz = Wz[s01 + i];               vh = Wh[s01 + i]; }
  else             { vz = Wz[s11 + i];               vh = Wh[s11 + i]; }
  pw[(m << 11) + i] = vz;
  pw[3 * 2048 + (m << 11) + i] = vh;
}

// ---------------------------------------------------------------- degree sums
__global__ void degrees_k(const int* __restrict__ row, const int* __restrict__ col,
                          const float* __restrict__ w, float* __restrict__ dout,
                          float* __restrict__ din) {
  int e = blockIdx.x * blockDim.x + threadIdx.x;
  if (e >= NE) return;
  float we = w[e];
  atomicAdd(&dout[row[e]], we);
  atomicAdd(&din[col[e]], we);
}

// ------------------------------------------- edge scatter: wave-per-edge, lane=channel
// Txo[c] += (w/deg_out[r]) * x[r];  Txi[r] += (w/deg_in[c]) * x[c]
__global__ void scatter_k(const int* __restrict__ row, const int* __restrict__ col,
                          const float* __restrict__ w, const float* __restrict__ x,
                          const float* __restrict__ dout, const float* __restrict__ din,
                          float* __restrict__ txo, float* __restrict__ txi) {
  unsigned tid = blockIdx.x * blockDim.x + threadIdx.x;
  unsigned e = tid >> 5;
  int ch = tid & 31;
  if (e >= NE) return;
  int r = row[e], c = col[e];
  float we = w[e];
  float no = we / dout[r];
  float ni = we / din[c];
  atomicAdd(&txo[c * CIN + ch], no * x[r * CIN + ch]);
  atomicAdd(&txi[r * CIN + ch], ni * x[c * CIN + ch]);
}

// ---------------------------------- fused WMMA GEMM + GRU gates + ReLU + Linear
// One wave per 16-node tile. f32 WMMA 16x16x4, K=32 in 8 steps, N=64 in 4 tiles,
// two gates (Z, H~). Fragment layouts per CDNA5 ISA 7.12.2:
//  A (16x4 f32): lanes 0-15 -> M=lane, {K=k0,k0+1}; lanes 16-31 -> {K=k0+2,k0+3}
//  B (4x16 f32): N = lane&15; v0/v1 hold rows K=k0+2*half / k0+1+2*half
//  C/D (16x16 f32): VGPR v -> M = v + 8*(lane>=16), N = lane&15
__global__ __launch_bounds__(256) void fused_wmma(
    const float* __restrict__ x, const float* __restrict__ txo,
    const float* __restrict__ txi, const float* __restrict__ pw,
    const float* __restrict__ bz, const float* __restrict__ bh,
    const float* __restrict__ Wlin, const float* __restrict__ blin,
    float* __restrict__ out) {
  const int lane = threadIdx.x & 31;
  const int wv   = threadIdx.x >> 5;
  const int tile = blockIdx.x * 8 + wv;         // wave-uniform
  if (tile >= NTILES) return;                   // whole-wave exit: EXEC stays full
  const int half  = lane >> 4;
  const int ln    = lane & 15;
  const int node0 = tile << 4;

  const float* __restrict__ srcs[3] = { x, txo, txi };

  v8f accZ[4] = {};
  v8f accH[4] = {};

  const int arow = (node0 + ln) * CIN + 2 * half;   // + ks*4 per k-step (8B aligned)

#pragma unroll
  for (int s = 0; s < 3; ++s) {
    const float* __restrict__ S   = srcs[s];
    const float* __restrict__ WzS = pw + s * 2048;
    const float* __restrict__ WhS = pw + 3 * 2048 + s * 2048;
#pragma unroll
    for (int ks = 0; ks < 8; ++ks) {
      v2f a = *(const v2f*)(S + arow + ks * 4);
      const int kb = (ks * 4 + 2 * half) * 64;
#pragma unroll
      for (int nt = 0; nt < 4; ++nt) {
        const int cidx = nt * 16 + ln;
        v2f bv;
        bv.x = WzS[kb + cidx];
        bv.y = WzS[kb + 64 + cidx];
        accZ[nt] = __builtin_amdgcn_wmma_f32_16x16x4_f32(
            false, a, false, bv, (short)0, accZ[nt], false, false);
        bv.x = WhS[kb + cidx];
        bv.y = WhS[kb + 64 + cidx];
        accH[nt] = __builtin_amdgcn_wmma_f32_16x16x4_f32(
            false, a, false, bv, (short)0, accH[nt], false, false);
      }
    }
  }

  // Epilogue: Z=sigmoid, H~=tanh, H=(1-Z)H~, relu, dot with Wlin, + blin
  float wl[4], bzf[4], bhf[4];
#pragma unroll
  for (int nt = 0; nt < 4; ++nt) {
    wl[nt]  = Wlin[nt * 16 + ln];
    bzf[nt] = bz[nt * 16 + ln];
    bhf[nt] = bh[nt * 16 + ln];
  }
  const float b0 = blin[0];

#pragma unroll
  for (int v = 0; v < 8; ++v) {
    float t = 0.f;
#pragma unroll
    for (int nt = 0; nt < 4; ++nt) {
      float zv = 1.f / (1.f + __expf(-(accZ[nt][v] + bzf[nt])));
      float hv = tanhf(accH[nt][v] + bhf[nt]);
      float h  = (1.f - zv) * hv;
      t += fmaxf(h, 0.f) * wl[nt];
    }
    // reduce over the 16 lanes of this half (xor of bits 0..3 stays in-half)
#pragma unroll
    for (int off = 8; off >= 1; off >>= 1)
      t += __shfl_xor(t, off, 32);
    if (ln == 0) out[node0 + v + 8 * half] = t + b0;
  }
}

// --------------------------------------------------------------------- launch
extern "C" void kernel_launch(void* const* d_in, const int* in_sizes, int n_in,
                              void* d_out, int out_size, void* d_ws, size_t ws_size,
                              hipStream_t stream) {
  const float* x    = (const float*)d_in[0];
  const int*   ei   = (const int*)d_in[1];
  const float* ew   = (const float*)d_in[2];
  const float* Wz   = (const float*)d_in[3];
  const float* bz   = (const float*)d_in[4];
  // d_in[5] = Wr, d_in[6] = br: dead (H0 == 0 makes the R gate a no-op)
  const float* Wh   = (const float*)d_in[7];
  const float* bh   = (const float*)d_in[8];
  const float* Wlin = (const float*)d_in[9];
  const float* blin = (const float*)d_in[10];

  const int* row = ei;        // edge_index[0]
  const int* col = ei + NE;   // edge_index[1]

  float* ws   = (float*)d_ws;
  float* dout = ws;                          // [NN]
  float* din  = ws + NN;                     // [NN]
  float* txo  = ws + 2 * NN;                 // [NN*32]
  float* txi  = ws + 2 * NN + NN * CIN;      // [NN*32]
  float* pw   = ws + 2 * NN + 2 * NN * CIN;  // [6*2048]

  const int nz = 2 * NN + 2 * NN * CIN;      // zero degrees + Tx accumulators
  zero_ws<<<(nz + 255) / 256, 256, 0, stream>>>(ws, nz);
  pack_w<<<(3 * 2048 + 255) / 256, 256, 0, stream>>>(Wz, Wh, pw);
  degrees_k<<<(NE + 255) / 256, 256, 0, stream>>>(row, col, ew, dout, din);

  const long total = (long)NE * 32;
  scatter_k<<<(int)((total + 255) / 256), 256, 0, stream>>>(
      row, col, ew, x, dout, din, txo, txi);

  fused_wmma<<<(NTILES + 7) / 8, 256, 0, stream>>>(
      x, txo, txi, pw, bz, bh, Wlin, blin, (float*)d_out);
}